// FeaStGNN_PrePool_36240934043738
// MI455X (gfx1250) — compile-verified
//
#include <hip/hip_runtime.h>
#include <cstdint>
#include <cstddef>

#define N1v 100000
#define N2v 25000
#define N3v 6250
#define E1v 600000
#define E2v 150000
#define E3v 37500
#define HEADS 6
#define SLOPE 0.1f

typedef float v2f __attribute__((ext_vector_type(2)));
typedef float v8f __attribute__((ext_vector_type(8)));

// ---------------------------------------------------------------------------
// Branch-free WMMA f32 GEMM: C[M,N] = A[M,K] @ B[K,N], row-major fp32.
// Preconditions (enforced by host-side padding): M%16==0, N%64==0, K%4==0.
// One wave computes a 16x64 strip: 4 WMMA accumulators share one A fragment.
// V_WMMA_F32_16X16X4_F32 operand layouts (ISA 7.12.2):
//   A (16x4): lane L holds row M=L%16; lanes 0-15 K={k0,k0+1}, 16-31 K={k0+2,k0+3}
//   B (4x16): VGPR v: lanes 0-15 row k0+2*half+v-pattern mirrored
//   C/D:      VGPR v: lanes 0-15 -> M=v, lanes 16-31 -> M=v+8
// ---------------------------------------------------------------------------
__global__ __launch_bounds__(256) void wmma_gemm_f32(const float* __restrict__ A,
                                                     const float* __restrict__ B,
                                                     float* __restrict__ C,
                                                     int M, int N, int K)
{
    const int lane   = threadIdx.x & 31;
    const int wave   = blockIdx.x * (blockDim.x >> 5) + (threadIdx.x >> 5);
    const int strips = N >> 6;                 // strips of 64 cols
    if (wave >= (M >> 4) * strips) return;
    const int tm   = (wave / strips) << 4;
    const int tn   = (wave % strips) << 6;
    const int half = lane >> 4;                // 0: lanes 0-15, 1: lanes 16-31
    const int l16  = lane & 15;

    const float* __restrict__ Arow = A + (size_t)(tm + l16) * K + (half << 1);
    const float* __restrict__ Bcol = B + tn + l16;

    v8f acc0 = {0.f,0.f,0.f,0.f,0.f,0.f,0.f,0.f};
    v8f acc1 = acc0, acc2 = acc0, acc3 = acc0;

#pragma unroll 2
    for (int k0 = 0; k0 < K; k0 += 4) {
        const v2f a = *(const v2f*)(Arow + k0);                     // 8B aligned
        const float* __restrict__ b0 = Bcol + (size_t)(k0 + (half << 1)) * N;
        const float* __restrict__ b1 = b0 + N;
        v2f bv0, bv1, bv2, bv3;
        bv0.x = b0[0];  bv0.y = b1[0];
        bv1.x = b0[16]; bv1.y = b1[16];
        bv2.x = b0[32]; bv2.y = b1[32];
        bv3.x = b0[48]; bv3.y = b1[48];
        acc0 = __builtin_amdgcn_wmma_f32_16x16x4_f32(false, a, false, bv0, (short)0, acc0, false, false);
        acc1 = __builtin_amdgcn_wmma_f32_16x16x4_f32(false, a, false, bv1, (short)0, acc1, false, false);
        acc2 = __builtin_amdgcn_wmma_f32_16x16x4_f32(false, a, false, bv2, (short)0, acc2, false, false);
        acc3 = __builtin_amdgcn_wmma_f32_16x16x4_f32(false, a, false, bv3, (short)0, acc3, false, false);
    }

    float* __restrict__ Cb = C + (size_t)(tm + (half << 3)) * N + tn + l16;
#pragma unroll
    for (int v = 0; v < 8; ++v) {
        float* cr = Cb + (size_t)v * N;
        cr[0]  = acc0[v];
        cr[16] = acc1[v];
        cr[32] = acc2[v];
        cr[48] = acc3[v];
    }
}

// Zero-padded copy A[M,K] -> Apad[Mpad,Kpad]
__global__ void pad_copy_kernel(const float* __restrict__ A, float* __restrict__ Apad,
                                int M, int K, int Mpad, int Kpad)
{
    const int idx = blockIdx.x * blockDim.x + threadIdx.x;
    if (idx >= Mpad * Kpad) return;
    const int i = idx / Kpad;
    const int k = idx % Kpad;
    Apad[idx] = (i < M && k < K) ? A[(size_t)i * K + k] : 0.f;
}

// P[i, 0..5] = X[i,:] @ u  (stride-8 padded rows)
__global__ void proj_u_kernel(const float* __restrict__ X, const float* __restrict__ u,
                              float* __restrict__ P, int n, int Cin)
{
    const int idx = blockIdx.x * blockDim.x + threadIdx.x;
    if (idx >= n * 8) return;
    const int i = idx >> 3;
    const int h = idx & 7;
    float v = 0.f;
    if (h < HEADS) {
        const float* xr = X + (size_t)i * Cin;
        for (int k = 0; k < Cin; ++k) v += xr[k] * u[k * HEADS + h];
    }
    P[idx] = v;
}

// One wave per edge: q = softmax(P[s]-P[d]+c); acc[d] += sum_h q_h * Y[s,h,:]
__global__ __launch_bounds__(256) void edge_msg_kernel(const int* __restrict__ src,
                                                       const int* __restrict__ dst,
                                                       const float* __restrict__ P,
                                                       const float* __restrict__ c,
                                                       const float* __restrict__ Y,
                                                       float* __restrict__ acc,
                                                       float* __restrict__ deg,
                                                       int E, int Cout, int HC)
{
    const int wid  = (blockIdx.x * blockDim.x + threadIdx.x) >> 5;
    const int lane = threadIdx.x & 31;
    if (wid >= E) return;
    const int s = src[wid];
    const int d = dst[wid];
    const float* Ps = P + (size_t)s * 8;
    const float* Pd = P + (size_t)d * 8;

    float l[HEADS];
    float mx = -1e30f;
#pragma unroll
    for (int h = 0; h < HEADS; ++h) { l[h] = Ps[h] - Pd[h] + c[h]; mx = fmaxf(mx, l[h]); }
    float ssum = 0.f;
#pragma unroll
    for (int h = 0; h < HEADS; ++h) { l[h] = expf(l[h] - mx); ssum += l[h]; }
    const float inv = 1.f / ssum;

    const float* ys = Y + (size_t)s * HC;
    float* ad = acc + (size_t)d * Cout;
    __builtin_prefetch((const void*)ys, 0, 1);   // global_prefetch_b8

    for (int j = lane; j < Cout; j += 32) {
        float v = 0.f;
#pragma unroll
        for (int h = 0; h < HEADS; ++h) v += l[h] * ys[h * Cout + j];
        atomicAdd(&ad[j], v * inv);
    }
    if (lane == 0) atomicAdd(&deg[d], 1.f);
}

// out[i, colOff+j] = (acc[i,j] + softmax(c)_h . Y[i,h,j]) / (deg[i]+1) + b[j]  (+lrelu)
__global__ void finalize_feast_kernel(const float* __restrict__ acc, const float* __restrict__ deg,
                                      const float* __restrict__ Y, const float* __restrict__ c,
                                      const float* __restrict__ b, float* __restrict__ out,
                                      int n, int Cout, int HC, int outStride, int colOff, int relu)
{
    const int idx = blockIdx.x * blockDim.x + threadIdx.x;
    if (idx >= n * Cout) return;
    const int i = idx / Cout;
    const int j = idx % Cout;

    float q[HEADS];
    float mx = -1e30f;
#pragma unroll
    for (int h = 0; h < HEADS; ++h) { q[h] = c[h]; mx = fmaxf(mx, q[h]); }
    float ssum = 0.f;
#pragma unroll
    for (int h = 0; h < HEADS; ++h) { q[h] = expf(q[h] - mx); ssum += q[h]; }
    const float inv = 1.f / ssum;

    const float* yi = Y + (size_t)i * HC;
    float selfm = 0.f;
#pragma unroll
    for (int h = 0; h < HEADS; ++h) selfm += q[h] * yi[h * Cout + j];

    float v = (acc[idx] + selfm * inv) / (deg[i] + 1.f) + b[j];
    if (relu) v = (v > 0.f) ? v : SLOPE * v;
    out[(size_t)i * outStride + colOff + j] = v;
}

__global__ void pool_scatter_kernel(const float* __restrict__ X, const int* __restrict__ cl,
                                    float* __restrict__ acc, int nf, int C)
{
    const int idx = blockIdx.x * blockDim.x + threadIdx.x;
    if (idx >= nf * C) return;
    const int i = idx / C;
    const int j = idx % C;
    atomicAdd(&acc[(size_t)cl[i] * C + j], X[idx]);
}

__global__ void pool_count_kernel(const int* __restrict__ cl, float* __restrict__ cnt, int nf)
{
    const int i = blockIdx.x * blockDim.x + threadIdx.x;
    if (i >= nf) return;
    atomicAdd(&cnt[cl[i]], 1.f);
}

__global__ void pool_div_kernel(const float* __restrict__ acc, const float* __restrict__ cnt,
                                float* __restrict__ out, int nc, int C)
{
    const int idx = blockIdx.x * blockDim.x + threadIdx.x;
    if (idx >= nc * C) return;
    const int i = idx / C;
    out[idx] = acc[idx] / fmaxf(cnt[i], 1.f);
}

__global__ void gather_rows_kernel(const float* __restrict__ src, const int* __restrict__ cl,
                                   float* __restrict__ dst, int n, int C)
{
    const int idx = blockIdx.x * blockDim.x + threadIdx.x;
    if (idx >= n * C) return;
    const int i = idx / C;
    const int j = idx % C;
    dst[idx] = src[(size_t)cl[i] * C + j];
}

__global__ void copy2d_kernel(const float* __restrict__ src, float* __restrict__ dst,
                              int n, int C, int dstStride, int colOff)
{
    const int idx = blockIdx.x * blockDim.x + threadIdx.x;
    if (idx >= n * C) return;
    const int i = idx / C;
    const int j = idx % C;
    dst[(size_t)i * dstStride + colOff + j] = src[idx];
}

__global__ void bias_act_kernel(float* __restrict__ X, const float* __restrict__ b, int n, int C)
{
    const int idx = blockIdx.x * blockDim.x + threadIdx.x;
    if (idx >= n * C) return;
    float v = X[idx] + b[idx % C];
    X[idx] = (v > 0.f) ? v : SLOPE * v;
}

// fused fc2 (512 -> 3) + L2 normalize; one wave per node
__global__ __launch_bounds__(256) void fc2_norm_kernel(const float* __restrict__ H,
                                                       const float* __restrict__ W2,
                                                       const float* __restrict__ b2,
                                                       float* __restrict__ out, int n)
{
    const int wid  = (blockIdx.x * blockDim.x + threadIdx.x) >> 5;
    const int lane = threadIdx.x & 31;
    if (wid >= n) return;
    const float* hr = H + (size_t)wid * 512;
    float s0 = 0.f, s1 = 0.f, s2 = 0.f;
    for (int k = lane; k < 512; k += 32) {
        const float hv = hr[k];
        s0 += hv * W2[k * 3 + 0];
        s1 += hv * W2[k * 3 + 1];
        s2 += hv * W2[k * 3 + 2];
    }
#pragma unroll
    for (int o = 16; o > 0; o >>= 1) {
        s0 += __shfl_down(s0, o, 32);
        s1 += __shfl_down(s1, o, 32);
        s2 += __shfl_down(s2, o, 32);
    }
    if (lane == 0) {
        s0 += b2[0]; s1 += b2[1]; s2 += b2[2];
        const float nrm = fmaxf(sqrtf(s0 * s0 + s1 * s1 + s2 * s2), 1e-12f);
        out[(size_t)wid * 3 + 0] = s0 / nrm;
        out[(size_t)wid * 3 + 1] = s1 / nrm;
        out[(size_t)wid * 3 + 2] = s2 / nrm;
    }
}

// ---------------------------------------------------------------------------
// Host-side orchestration
// ---------------------------------------------------------------------------
static inline int cdiv(int a, int b) { return (a + b - 1) / b; }

// Requires N % 64 == 0 (true for all layers: 192/384/768/512).
// Pads A (via Apad scratch) so M%16==0 and K%4==0; C must have Mpad rows.
static void launch_gemm(const float* A, const float* B, float* C, int M, int N, int K,
                        float* Apad, hipStream_t s)
{
    const int Mp = (M + 15) & ~15;
    const int Kp = (K + 3) & ~3;
    const float* Ause = A;
    if (Mp != M || Kp != K) {
        pad_copy_kernel<<<cdiv(Mp * Kp, 256), 256, 0, s>>>(A, Apad, M, K, Mp, Kp);
        Ause = Apad;
    }
    const int waves = (Mp >> 4) * (N >> 6);
    wmma_gemm_f32<<<cdiv(waves, 8), 256, 0, s>>>(Ause, B, C, Mp, N, Kp);
}

struct FeastP { const float *W, *u, *c, *b; };

static void run_feast(const float* X, const int* src, const int* dst, int E,
                      const FeastP& p, float* out, int outStride, int colOff, int relu,
                      int n, int Cin, int Cout,
                      float* Y, float* P, float* acc, float* deg, float* Apad, hipStream_t s)
{
    const int HC = HEADS * Cout;
    proj_u_kernel<<<cdiv(n * 8, 256), 256, 0, s>>>(X, p.u, P, n, Cin);
    launch_gemm(X, p.W, Y, n, HC, Cin, Apad, s);
    hipMemsetAsync(acc, 0, (size_t)n * Cout * sizeof(float), s);
    hipMemsetAsync(deg, 0, (size_t)n * sizeof(float), s);
    edge_msg_kernel<<<cdiv(E, 8), 256, 0, s>>>(src, dst, P, p.c, Y, acc, deg, E, Cout, HC);
    finalize_feast_kernel<<<cdiv(n * Cout, 256), 256, 0, s>>>(acc, deg, Y, p.c, p.b, out,
                                                              n, Cout, HC, outStride, colOff, relu);
}

static void run_pool(const float* X, const int* cl, float* out, int nf, int nc, int C,
                     float* acc, float* cnt, hipStream_t s)
{
    hipMemsetAsync(acc, 0, (size_t)nc * C * sizeof(float), s);
    hipMemsetAsync(cnt, 0, (size_t)nc * sizeof(float), s);
    pool_scatter_kernel<<<cdiv(nf * C, 256), 256, 0, s>>>(X, cl, acc, nf, C);
    pool_count_kernel<<<cdiv(nf, 256), 256, 0, s>>>(cl, cnt, nf);
    pool_div_kernel<<<cdiv(nc * C, 256), 256, 0, s>>>(acc, cnt, out, nc, C);
}

extern "C" void kernel_launch(void* const* d_in, const int* in_sizes, int n_in,
                              void* d_out, int out_size, void* d_ws, size_t ws_size,
                              hipStream_t stream)
{
    (void)in_sizes; (void)n_in; (void)out_size; (void)ws_size;

    const float* x   = (const float*)d_in[0];
    const int*   ei1 = (const int*)d_in[1];
    const int*   ei2 = (const int*)d_in[2];
    const int*   ei3 = (const int*)d_in[3];
    const int*   cl1 = (const int*)d_in[4];
    const int*   cl2 = (const int*)d_in[5];

    int pi = 6;
    auto nextF = [&]() {
        FeastP f;
        f.W = (const float*)d_in[pi++]; f.u = (const float*)d_in[pi++];
        f.c = (const float*)d_in[pi++]; f.b = (const float*)d_in[pi++];
        return f;
    };
    const FeastP conv1 = nextF(), conv2 = nextF(), conv3 = nextF(), conv4 = nextF();
    const FeastP lin5  = nextF(), conv5 = nextF(), lin6  = nextF(), conv6 = nextF();
    const float* fc1W = (const float*)d_in[pi++];
    const float* fc1b = (const float*)d_in[pi++];
    const float* fc2W = (const float*)d_in[pi++];
    const float* fc2b = (const float*)d_in[pi++];

    // ---- workspace carve-out (floats) ----
    float* ws = (float*)d_ws;
    size_t off = 0;
    auto alloc = [&](size_t nfl) { float* p = ws + off; off += (nfl + 63) & ~(size_t)63; return p; };
    const size_t N2p = 25008, N3p = 6256;     // round16 of N2/N3 (N1=100000 is 6250*16)
    float* x1    = alloc((size_t)N1v * 32);   // lrelu(conv1)
    float* x2a   = alloc((size_t)N2v * 32);   // pool1
    float* x2    = alloc((size_t)N2v * 64);   // lrelu(conv2)
    float* x3a   = alloc((size_t)N3v * 64);   // pool2
    float* x3    = alloc((size_t)N3v * 128);  // lrelu(conv3)
    float* x3b   = alloc((size_t)N3v * 128);  // lrelu(conv4)
    float* x3g   = alloc((size_t)N2v * 128);  // x3b[cluster2]
    float* f2d   = alloc((size_t)N2v * 64);   // lin5 (no act)
    float* x2cat = alloc((size_t)N2v * 128);  // concat(x2, f2d)
    float* x2b   = alloc((size_t)N2v * 64);   // lrelu(conv5)
    float* x2g   = alloc((size_t)N1v * 64);   // x2b[cluster1]
    float* f1d   = alloc((size_t)N1v * 32);   // lin6 (no act)
    float* x1cat = alloc((size_t)N1v * 64);   // concat(x1, f1d)
    float* ff    = alloc((size_t)N1v * 32);   // lrelu(conv6)
    float* h1    = alloc((size_t)N1v * 512);  // lrelu(fc1)
    float* Y     = alloc((size_t)N1v * 192);  // per-layer X@W scratch (holds padded rows too)
    float* P     = alloc((size_t)N1v * 8);    // per-layer X@u scratch (stride 8)
    float* acc   = alloc((size_t)N1v * 32);   // per-layer aggregation scratch (max size)
    float* deg   = alloc((size_t)N1v);        // per-layer degree / pool counts
    float* Apad  = alloc(N2p * 128);          // zero-padded GEMM A operand (max padded case)

    float* out = (float*)d_out;

    // ---- pipeline ----
    run_feast(x, ei1, ei1 + E1v, E1v, conv1, x1, 32, 0, 1, N1v, 3, 32, Y, P, acc, deg, Apad, stream);
    run_pool(x1, cl1, x2a, N1v, N2v, 32, acc, deg, stream);
    run_feast(x2a, ei2, ei2 + E2v, E2v, conv2, x2, 64, 0, 1, N2v, 32, 64, Y, P, acc, deg, Apad, stream);
    run_pool(x2, cl2, x3a, N2v, N3v, 64, acc, deg, stream);
    run_feast(x3a, ei3, ei3 + E3v, E3v, conv3, x3, 128, 0, 1, N3v, 64, 128, Y, P, acc, deg, Apad, stream);
    run_feast(x3, ei3, ei3 + E3v, E3v, conv4, x3b, 128, 0, 1, N3v, 128, 128, Y, P, acc, deg, Apad, stream);

    gather_rows_kernel<<<cdiv(N2v * 128, 256), 256, 0, stream>>>(x3b, cl2, x3g, N2v, 128);
    run_feast(x3g, ei2, ei2 + E2v, E2v, lin5, f2d, 64, 0, 0, N2v, 128, 64, Y, P, acc, deg, Apad, stream);
    copy2d_kernel<<<cdiv(N2v * 64, 256), 256, 0, stream>>>(x2, x2cat, N2v, 64, 128, 0);
    copy2d_kernel<<<cdiv(N2v * 64, 256), 256, 0, stream>>>(f2d, x2cat, N2v, 64, 128, 64);
    run_feast(x2cat, ei2, ei2 + E2v, E2v, conv5, x2b, 64, 0, 1, N2v, 128, 64, Y, P, acc, deg, Apad, stream);

    gather_rows_kernel<<<cdiv(N1v * 64, 256), 256, 0, stream>>>(x2b, cl1, x2g, N1v, 64);
    run_feast(x2g, ei1, ei1 + E1v, E1v, lin6, f1d, 32, 0, 0, N1v, 64, 32, Y, P, acc, deg, Apad, stream);
    copy2d_kernel<<<cdiv(N1v * 32, 256), 256, 0, stream>>>(x1, x1cat, N1v, 32, 64, 0);
    copy2d_kernel<<<cdiv(N1v * 32, 256), 256, 0, stream>>>(f1d, x1cat, N1v, 32, 64, 32);
    run_feast(x1cat, ei1, ei1 + E1v, E1v, conv6, ff, 32, 0, 1, N1v, 64, 32, Y, P, acc, deg, Apad, stream);

    // fc1 via WMMA (M=100000, N=512, K=32 all exact), then bias + leaky relu
    launch_gemm(ff, fc1W, h1, N1v, 512, 32, Apad, stream);
    bias_act_kernel<<<cdiv(N1v * 512, 256), 256, 0, stream>>>(h1, fc1b, N1v, 512);

    // fused fc2 + L2 normalize
    fc2_norm_kernel<<<cdiv(N1v, 8), 256, 0, stream>>>(h1, fc2W, fc2b, out, N1v);
}